// ThetaModel_45552423141376
// MI455X (gfx1250) — compile-verified
//
#include <hip/hip_runtime.h>

typedef __attribute__((ext_vector_type(2))) float v2f;
typedef __attribute__((ext_vector_type(8))) float v8f;

#define M_TOTAL   524288
#define BLOCKS    512
#define THREADS   256
#define WAVES_TOT (BLOCKS * (THREADS / 32))          // 4096 waves
#define M_PER_WAVE (M_TOTAL / WAVES_TOT)             // 128 rows per wave

// Seed output with the bias h (shape (2,16,16) flat == 512 floats, same order as out).
__global__ void theta_init_kernel(const float* __restrict__ h, float* __restrict__ out) {
    int i = blockIdx.x * blockDim.x + threadIdx.x;
    if (i < 512) out[i] = h[i];
}

__global__ __launch_bounds__(THREADS) void theta_wmma_kernel(
    const float* __restrict__ gi,    // (2, M, 16):  e = plane 0, f = plane 1
    const float* __restrict__ gj,    // (2, M, 16):  a = plane 0, b = plane 1
    const float* __restrict__ theta, // (2, M)
    float* __restrict__ out)         // (2, 16, 16)
{
    const int lane  = threadIdx.x & 31;
    const int khalf = lane >> 4;       // 0: K in {0,1}, 1: K in {2,3}
    const int col   = lane & 15;       // M-row (for A) / N-col (for B,C,D)
    const int wave  = blockIdx.x * (THREADS / 32) + (threadIdx.x >> 5);

    const float* __restrict__ e_base = gi;
    const float* __restrict__ f_base = gi + (size_t)M_TOTAL * 16;
    const float* __restrict__ a_base = gj;
    const float* __restrict__ b_base = gj + (size_t)M_TOTAL * 16;
    const float* __restrict__ th0    = theta;
    const float* __restrict__ th1    = theta + M_TOTAL;

    v8f acc_re = {};
    v8f acc_im = {};

    const int m0 = wave * M_PER_WAVE;

    for (int it = 0; it < M_PER_WAVE; it += 4) {
        const int mA = m0 + it + 2 * khalf;   // row feeding VGPR0 of A/B
        const int mB = mA + 1;                // row feeding VGPR1 of A/B
        const size_t offA = (size_t)mA * 16 + col;
        const size_t offB = (size_t)mB * 16 + col;

        // A matrices: 16x4 f32, lane = M-row, VGPR v = K row (2*khalf + v)
        v2f Ae = { e_base[offA], e_base[offB] };
        v2f Af = { f_base[offA], f_base[offB] };

        // Rotation inputs
        const float a0 = a_base[offA], a1 = a_base[offB];
        const float b0 = b_base[offA], b1 = b_base[offB];
        const float c0 = __cosf(th0[mA]);
        const float c1 = __cosf(th0[mB]);
        const float s0 = __sinf(th1[mA]);
        const float s1 = __sinf(th1[mB]);

        // B matrices: 4x16 f32, lane = N-col, VGPR v = K row (2*khalf + v)
        v2f Bre  = { a0 * c0 - b0 * s0, a1 * c1 - b1 * s1 };   // u_re
        v2f Bim  = { a0 * s0 + b0 * c0, a1 * s1 + b1 * c1 };   // u_im
        v2f Bnim = { -Bim[0], -Bim[1] };                       // -u_im (F32 WMMA has no A/B neg)

        // out_re += e^T u_re - f^T u_im ; out_im += e^T u_im + f^T u_re
        acc_re = __builtin_amdgcn_wmma_f32_16x16x4_f32(false, Ae, false, Bre,  (short)0, acc_re, false, false);
        acc_re = __builtin_amdgcn_wmma_f32_16x16x4_f32(false, Af, false, Bnim, (short)0, acc_re, false, false);
        acc_im = __builtin_amdgcn_wmma_f32_16x16x4_f32(false, Ae, false, Bim,  (short)0, acc_im, false, false);
        acc_im = __builtin_amdgcn_wmma_f32_16x16x4_f32(false, Af, false, Bre,  (short)0, acc_im, false, false);
    }

    // Block-level reduction of the 8 per-block waves via LDS float atomics,
    // then one global float atomic per element per block.
    __shared__ float sred[512];
    for (int i = threadIdx.x; i < 512; i += THREADS) sred[i] = 0.0f;
    __syncthreads();

    #pragma unroll
    for (int r = 0; r < 8; ++r) {
        const int row = r + 8 * khalf;                 // C/D layout: VGPR r -> M row
        atomicAdd(&sred[row * 16 + col],       acc_re[r]);   // ds_add_f32
        atomicAdd(&sred[256 + row * 16 + col], acc_im[r]);
    }
    __syncthreads();

    for (int i = threadIdx.x; i < 512; i += THREADS) {
        unsafeAtomicAdd(&out[i], sred[i]);             // global_atomic_add_f32
    }
}

extern "C" void kernel_launch(void* const* d_in, const int* in_sizes, int n_in,
                              void* d_out, int out_size, void* d_ws, size_t ws_size,
                              hipStream_t stream) {
    const float* h     = (const float*)d_in[0];   // (2,1,1,16,16) -> 512
    const float* gi    = (const float*)d_in[1];   // (2,M,1,16)
    const float* gj    = (const float*)d_in[2];   // (2,1,M,16)
    const float* theta = (const float*)d_in[3];   // (2,M)
    float* out = (float*)d_out;                   // (2,16,16) -> 512

    theta_init_kernel<<<2, 256, 0, stream>>>(h, out);
    theta_wmma_kernel<<<BLOCKS, THREADS, 0, stream>>>(gi, gj, theta, out);
}